// ModelNew_25056839204968
// MI455X (gfx1250) — compile-verified
//
#include <hip/hip_runtime.h>
#include <hip/hip_bf16.h>

typedef float v2f __attribute__((ext_vector_type(2)));
typedef float v8f __attribute__((ext_vector_type(8)));

// Problem constants
// x:  [16, 16, 16, 32, 32]  (N, Cin, D, H, W)
// w:  [16, 32, 3, 3, 3]     (Cin, Cout, kD, kH, kW)
// y:  [16, 32, 31, 63, 63]
#define SPATIAL 123039   // 31*63*63
#define HW_OUT  3969     // 63*63
#define CI_STRIDE 16384  // 16*32*32 floats between consecutive Cin planes

// ---------------------------------------------------------------------------
// Fully-unrolled per-parity-class accumulator. RD/RH/RW are the output
// parities; tap sets are compile-time so all loops unroll into a straight
// line of V_WMMA_F32_16X16X4_F32 ops (4..32 per call).
//   A (M=16 couts, K)  from LDS;  B (K, N=16 output columns) from global.
//   Fragment layout (ISA 7.12.2): k = vgpr + 2*(lane>=16), m/n = lane%16.
// ---------------------------------------------------------------------------
template <int RD, int RH, int RW>
__device__ inline v8f conv_accum(const float* __restrict__ x,
                                 const float* __restrict__ Wlds,
                                 int n, int odp, int ohp, int owbase,
                                 int cobase, int hi, int lo) {
  constexpr int NTD = RD ? 2 : 1;
  constexpr int NTH = RH ? 2 : 1;
  constexpr int NTW = RW ? 2 : 1;
  v8f acc = {0.f, 0.f, 0.f, 0.f, 0.f, 0.f, 0.f, 0.f};

#pragma unroll
  for (int td = 0; td < NTD; ++td) {
    const int kd = RD ? (td == 0 ? 0 : 2) : 1;
    const int dd = RD ? (td == 0 ? 1 : 0) : 0;
#pragma unroll
    for (int th = 0; th < NTH; ++th) {
      const int kh = RH ? (th == 0 ? 0 : 2) : 1;
      const int dh = RH ? (th == 0 ? 1 : 0) : 0;
#pragma unroll
      for (int tw = 0; tw < NTW; ++tw) {
        const int kw = RW ? (tw == 0 ? 0 : 2) : 1;
        const int dw = RW ? (tw == 0 ? 1 : 0) : 0;
        const int kidx = (kd * 3 + kh) * 3 + kw;
        const int id = odp + dd;
        const int ih = ohp + dh;
        int iw = owbase + lo + dw;
        iw = iw > 31 ? 31 : iw;  // clamp; that column is masked at store

        const float* xp = x + (((n * 16) * 16 + id) * 32 + ih) * 32 + iw;
        const float* wl = Wlds + kidx * 512 + cobase + lo;

#pragma unroll
        for (int cb = 0; cb < 16; cb += 4) {  // K = 16 Cin per tap
          const int k0 = cb + (hi << 1);
          v2f a, b;
          a.x = wl[k0 * 32];
          a.y = wl[(k0 + 1) * 32];
          b.x = xp[k0 * CI_STRIDE];
          b.y = xp[(k0 + 1) * CI_STRIDE];
          acc = __builtin_amdgcn_wmma_f32_16x16x4_f32(
              false, a, false, b, (short)0, acc, false, false);
        }
      }
    }
  }
  return acc;
}

// ---------------------------------------------------------------------------
// Kernel 1: transposed conv via parity decomposition + WMMA.
// Block = (n, rd, rh, odp, ohp, ow-tile); 4 waves = {rw parity} x {cout half}.
// ---------------------------------------------------------------------------
__global__ __launch_bounds__(128) void convt_wmma_kernel(
    const float* __restrict__ x, const float* __restrict__ w,
    float* __restrict__ out)
{
  __shared__ float Wlds[27 * 16 * 32];  // [kidx][ci][co]

  int bid = blockIdx.x;
  const int wt  = bid & 1;  bid >>= 1;   // ow tile (0..1)
  const int ohp = bid & 31; bid >>= 5;   // h position within parity class
  const int odp = bid & 15; bid >>= 4;   // d position within parity class
  const int rh  = bid & 1;  bid >>= 1;   // h parity
  const int rd  = bid & 1;  bid >>= 1;   // d parity
  const int n   = bid;                   // batch

  if (odp >= 16 - rd || ohp >= 32 - rh) return;  // uniform early-exit

  // Stage weights into LDS transposed: Wlds[(kidx*16+ci)*32 + co]
  for (int i = threadIdx.x; i < 27 * 16 * 32; i += 128) {
    const int co   = i & 31;
    const int t    = i >> 5;
    const int ci   = t & 15;
    const int kidx = t >> 4;
    Wlds[i] = w[(ci * 32 + co) * 27 + kidx];
  }
  __syncthreads();

  // Wave id as an SGPR so rw/cobase are scalar (no EXEC-masked WMMA regions).
  const int wid    = __builtin_amdgcn_readfirstlane(threadIdx.x >> 5);
  const int lane   = threadIdx.x & 31;
  const int hi     = lane >> 4;       // lane group (0/1)
  const int lo     = lane & 15;       // m/n index within fragment
  const int rw     = wid >> 1;        // w parity handled by this wave (scalar)
  const int cobase = (wid & 1) << 4;  // cout half (scalar)
  const int owbase = wt << 4;

  v8f acc;
  const int key = (rd << 2) | (rh << 1) | rw;  // all scalar -> scalar branches
  switch (key) {
    case 0: acc = conv_accum<0,0,0>(x, Wlds, n, odp, ohp, owbase, cobase, hi, lo); break;
    case 1: acc = conv_accum<0,0,1>(x, Wlds, n, odp, ohp, owbase, cobase, hi, lo); break;
    case 2: acc = conv_accum<0,1,0>(x, Wlds, n, odp, ohp, owbase, cobase, hi, lo); break;
    case 3: acc = conv_accum<0,1,1>(x, Wlds, n, odp, ohp, owbase, cobase, hi, lo); break;
    case 4: acc = conv_accum<1,0,0>(x, Wlds, n, odp, ohp, owbase, cobase, hi, lo); break;
    case 5: acc = conv_accum<1,0,1>(x, Wlds, n, odp, ohp, owbase, cobase, hi, lo); break;
    case 6: acc = conv_accum<1,1,0>(x, Wlds, n, odp, ohp, owbase, cobase, hi, lo); break;
    default: acc = conv_accum<1,1,1>(x, Wlds, n, odp, ohp, owbase, cobase, hi, lo); break;
  }

  // Store D: co = cobase + r + 8*hi, output column = owbase + lo
  const int owp = owbase + lo;
  if (owp < 32 - rw) {
    const int od = 2 * odp + rd, oh = 2 * ohp + rh, ow = 2 * owp + rw;
    const long sp = (long)od * HW_OUT + oh * 63 + ow;
#pragma unroll
    for (int r = 0; r < 8; ++r) {
      const int co = cobase + (hi << 3) + r;
      out[(long)(n * 32 + co) * SPATIAL + sp] = acc[r];
    }
  }
}

// ---------------------------------------------------------------------------
// Kernel 2: per-(n,c) sum and sum-of-squares over the 123039 spatial elems
// ---------------------------------------------------------------------------
__global__ __launch_bounds__(256) void reduce_nc_kernel(
    const float* __restrict__ y, double* __restrict__ sums,
    double* __restrict__ sumsq)
{
  __shared__ double ls[256];
  __shared__ double lq[256];
  const int nc = blockIdx.x;  // 0..511
  const float* p = y + (long)nc * SPATIAL;
  double s = 0.0, q = 0.0;
  for (int i = threadIdx.x; i < SPATIAL; i += 256) {
    const double v = (double)p[i];
    s += v;
    q += v * v;
  }
  ls[threadIdx.x] = s;
  lq[threadIdx.x] = q;
  __syncthreads();
  for (int off = 128; off > 0; off >>= 1) {
    if (threadIdx.x < (unsigned)off) {
      ls[threadIdx.x] += ls[threadIdx.x + off];
      lq[threadIdx.x] += lq[threadIdx.x + off];
    }
    __syncthreads();
  }
  if (threadIdx.x == 0) { sums[nc] = ls[0]; sumsq[nc] = lq[0]; }
}

// ---------------------------------------------------------------------------
// Kernel 3: channel scale = gamma * rsqrt(var+eps); per-(n,c) spatial means.
// (conv bias, BN mean and BN bias cancel algebraically.)
// ---------------------------------------------------------------------------
__global__ void finalize_stats_kernel(
    const double* __restrict__ sums, const double* __restrict__ sumsq,
    const float* __restrict__ gamma, float* __restrict__ scale,
    float* __restrict__ mnc)
{
  const int c = threadIdx.x;  // 32 threads
  double s = 0.0, q = 0.0;
  for (int nn = 0; nn < 16; ++nn) {
    const double sn = sums[nn * 32 + c];
    s += sn;
    q += sumsq[nn * 32 + c];
    mnc[nn * 32 + c] = (float)(sn * (1.0 / (double)SPATIAL));
  }
  const double cnt  = 16.0 * (double)SPATIAL;
  const double mean = s / cnt;
  const double var  = q / cnt - mean * mean;
  scale[c] = gamma[c] * (float)(1.0 / sqrt(var + 1e-5));
}

// ---------------------------------------------------------------------------
// Kernel 4: in-place  y = scale[c] * (y - mean_spatial[n,c])
// ---------------------------------------------------------------------------
__global__ __launch_bounds__(256) void normalize_kernel(
    float* __restrict__ y, const float* __restrict__ scale,
    const float* __restrict__ mnc)
{
  const int nc = blockIdx.y;
  const float sc = scale[nc & 31];
  const float m  = mnc[nc];
  const int i = blockIdx.x * 256 + threadIdx.x;
  if (i < SPATIAL) {
    float* p = y + (long)nc * SPATIAL;
    p[i] = sc * (p[i] - m);
  }
}

extern "C" void kernel_launch(void* const* d_in, const int* in_sizes, int n_in,
                              void* d_out, int out_size, void* d_ws,
                              size_t ws_size, hipStream_t stream) {
  (void)in_sizes; (void)n_in; (void)out_size; (void)ws_size;
  const float* x     = (const float*)d_in[0];
  const float* w     = (const float*)d_in[1];
  // d_in[2] (conv_bias) and d_in[4] (bn_bias) cancel algebraically.
  const float* gamma = (const float*)d_in[3];
  float* out = (float*)d_out;

  double* sums  = (double*)d_ws;           // 512 doubles
  double* sumsq = sums + 512;              // 512 doubles
  float*  scale = (float*)(sumsq + 512);   // 32 floats
  float*  mnc   = scale + 32;              // 512 floats

  // grid: n(32) x rd(2) x rh(2) x odp(16) x ohp(32) x owtile(2) = 131072
  convt_wmma_kernel<<<131072, 128, 0, stream>>>(x, w, out);
  reduce_nc_kernel<<<512, 256, 0, stream>>>(out, sums, sumsq);
  finalize_stats_kernel<<<1, 32, 0, stream>>>(sums, sumsq, gamma, scale, mnc);
  normalize_kernel<<<dim3((SPATIAL + 255) / 256, 512), 256, 0, stream>>>(
      out, scale, mnc);
}